// SAGE_NET_39092792328709
// MI455X (gfx1250) — compile-verified
//
#include <hip/hip_runtime.h>

typedef float v2f __attribute__((ext_vector_type(2)));
typedef float v8f __attribute__((ext_vector_type(8)));

#define N_NODES 50000
#define DIM     128
#define LDSROW  132   // 128 + 4 pad: row base bank = 4*m -> conflict-free b64 frag loads

// ---------------- zero workspace ----------------
__global__ void k_zero(float4* __restrict__ agg4, float* __restrict__ deg,
                       int n4, int nd) {
    int i = blockIdx.x * blockDim.x + threadIdx.x;
    if (i < n4) agg4[i] = make_float4(0.f, 0.f, 0.f, 0.f);
    if (i < nd) deg[i] = 0.f;
}

// ---------------- edge scatter: wave per edge ----------------
__global__ void k_scatter(const float* __restrict__ x,
                          const long long* __restrict__ ei,
                          float* __restrict__ agg_sum,
                          float* __restrict__ deg, int E) {
    int wave = (int)((blockIdx.x * (unsigned)blockDim.x + threadIdx.x) >> 5);
    int lane = threadIdx.x & 31;
    if (wave >= E) return;
    long long s = ei[wave];       // src
    long long d = ei[(size_t)E + wave];   // dst
    float4 v = *(const float4*)(x + (size_t)s * DIM + lane * 4);
    float* dp = agg_sum + (size_t)d * DIM + lane * 4;
    atomicAdd(dp + 0, v.x);
    atomicAdd(dp + 1, v.y);
    atomicAdd(dp + 2, v.z);
    atomicAdd(dp + 3, v.w);
    if (lane == 0) atomicAdd(&deg[d], 1.0f);
}

// ---------------- fused mean + dual GEMM + bias + ReLU ----------------
// block = 256 threads (8 waves). Block handles 16 node rows x 128 output cols.
// Wave w computes the 16x16 tile at cols [16w, 16w+16) via v_wmma_f32_16x16x4_f32.
__global__ __launch_bounds__(256) void k_sage_gemm(
    const float* __restrict__ x, const float* __restrict__ agg_sum,
    const float* __restrict__ deg,
    const float* __restrict__ Wl, const float* __restrict__ bl,
    const float* __restrict__ Wr, float* __restrict__ out) {
    __shared__ float sA[16 * LDSROW];  // mean-aggregated neighbor features
    __shared__ float sX[16 * LDSROW];  // root features

    const int row0 = blockIdx.x * 16;
    const int t = threadIdx.x;

    // Stage 16x128 of agg (scaled by 1/max(deg,1)) and x into LDS, coalesced.
    {
        const int lr = t >> 5;          // 0..7
        const int cc = (t & 31) * 4;    // 0..124
#pragma unroll
        for (int it = 0; it < 2; ++it) {
            const int r = lr + it * 8;
            const int g = row0 + r;
            const float dg = deg[g];
            const float sc = 1.0f / fmaxf(dg, 1.0f);
            float4 a  = *(const float4*)(agg_sum + (size_t)g * DIM + cc);
            float4 xr = *(const float4*)(x       + (size_t)g * DIM + cc);
            a.x *= sc; a.y *= sc; a.z *= sc; a.w *= sc;
            *(float4*)(&sA[r * LDSROW + cc]) = a;
            *(float4*)(&sX[r * LDSROW + cc]) = xr;
        }
    }
    __syncthreads();

    const int wv   = t >> 5;       // wave id -> column tile
    const int lane = t & 31;
    const int col0 = wv * 16;
    const int m    = lane & 15;            // A row / B col / bias-col index
    const int kb   = (lane >> 4) * 2;      // K sub-offset from lane half

    const float* wl = Wl + (size_t)(col0 + m) * DIM;  // row of W_l -> B fragment
    const float* wr = Wr + (size_t)(col0 + m) * DIM;

    v8f acc = {};
    // out += agg @ W_l^T
#pragma unroll
    for (int k0 = 0; k0 < DIM; k0 += 4) {
        v2f a = *(const v2f*)(&sA[m * LDSROW + k0 + kb]);
        v2f b = *(const v2f*)(&wl[k0 + kb]);
        acc = __builtin_amdgcn_wmma_f32_16x16x4_f32(false, a, false, b,
                                                    (short)0, acc, false, false);
    }
    // out += x @ W_r^T
#pragma unroll
    for (int k0 = 0; k0 < DIM; k0 += 4) {
        v2f a = *(const v2f*)(&sX[m * LDSROW + k0 + kb]);
        v2f b = *(const v2f*)(&wr[k0 + kb]);
        acc = __builtin_amdgcn_wmma_f32_16x16x4_f32(false, a, false, b,
                                                    (short)0, acc, false, false);
    }

    const float bias  = bl[col0 + m];
    const int   rbase = row0 + ((lane >> 4) << 3);
#pragma unroll
    for (int i = 0; i < 8; ++i) {
        float vo = acc[i] + bias;
        vo = fmaxf(vo, 0.0f);
        out[(size_t)(rbase + i) * DIM + col0 + m] = vo;
    }
}

extern "C" void kernel_launch(void* const* d_in, const int* in_sizes, int n_in,
                              void* d_out, int out_size, void* d_ws, size_t ws_size,
                              hipStream_t stream) {
    const float*     x  = (const float*)d_in[0];
    const long long* ei = (const long long*)d_in[1];
    const float*     Wl = (const float*)d_in[2];
    const float*     bl = (const float*)d_in[3];
    const float*     Wr = (const float*)d_in[4];
    float*           out = (float*)d_out;

    float* agg_sum = (float*)d_ws;                       // N*128 f32
    float* deg     = agg_sum + (size_t)N_NODES * DIM;    // N f32

    const int E = in_sizes[1] / 2;

    // 1) zero accumulators
    const int n4 = N_NODES * DIM / 4;
    k_zero<<<(n4 + 255) / 256, 256, 0, stream>>>((float4*)agg_sum, deg, n4, N_NODES);

    // 2) scatter-sum + degree (wave per edge)
    {
        const long long waves = E;
        const int blocks = (int)((waves * 32 + 255) / 256);
        k_scatter<<<blocks, 256, 0, stream>>>(x, ei, agg_sum, deg, E);
    }

    // 3) fused mean + GEMMs + bias + ReLU  (N_NODES == 3125 * 16 exactly)
    k_sage_gemm<<<N_NODES / 16, 256, 0, stream>>>(x, agg_sum, deg, Wl, bl, Wr, out);
}